// HMM_50826642981336
// MI455X (gfx1250) — compile-verified
//
#include <hip/hip_runtime.h>
#include <stdint.h>

// Problem constants (match reference)
#define B_DIM 256
#define T_DIM 4096
#define S_DIM 32          // == wave32: lane <-> state
#define L_CHUNK 128       // timesteps per chunk
#define NCHUNK (T_DIM / L_CHUNK)   // 32 chunks per sequence

#define NEG_HALF_LN2PI (-0.91893853320467274178f)

typedef __attribute__((ext_vector_type(4))) unsigned int uint32x4;
typedef __attribute__((ext_vector_type(8))) int          int32x8;
typedef __attribute__((ext_vector_type(4))) int          int32x4;

__device__ __forceinline__ float emit_ll(float x, float mu, float inv_sig, float ls) {
    // -0.5*z^2 - log_sigma - 0.5*ln(2*pi),  z = (x - mu) * exp(-log_sigma)
    float z = (x - mu) * inv_sig;
    return fmaf(-0.5f * z, z, NEG_HALF_LN2PI - ls);
}

// Build a 1-D TDM descriptor (n contiguous 4-byte elements, global <-> LDS).
__device__ __forceinline__ void tdm_desc_1d(uint64_t ga, uint32_t la, uint32_t n,
                                            uint32x4& g0, int32x8& g1) {
    g0[0] = 1u;                                   // count = 1 valid descriptor
    g0[1] = la;                                   // lds_addr (bytes)
    g0[2] = (uint32_t)ga;                         // global_addr[31:0]
    g0[3] = (uint32_t)(ga >> 32) | (2u << 30);    // global_addr[56:32] | type=2
    g1[0] = (int)(2u << 16);                      // wg_mask=0 | data_size=2 (4 bytes)
    g1[1] = (int)((n & 0xFFFFu) << 16);           // tensor_dim0[15:0] @ bits63:48
    g1[2] = (int)((n >> 16) | (1u << 16));        // tensor_dim0[31:16] | tensor_dim1 = 1
    g1[3] = (int)((n & 0xFFFFu) << 16);           // tile_dim0 = n @ bits127:112
    g1[4] = (int)1u;                              // tile_dim1 = 1, tile_dim2 = 0
    g1[5] = (int)n;                               // tensor_dim0_stride = n
    g1[6] = 0;
    g1[7] = 0;
}

__device__ __forceinline__ void tdm_load_1d(const void* gptr, const void* lptr, uint32_t n) {
    uint32x4 g0; int32x8 g1;
    tdm_desc_1d((uint64_t)(uintptr_t)gptr, (uint32_t)(uintptr_t)lptr, n, g0, g1);
    int32x4 gz4 = {0, 0, 0, 0};
    int32x8 gz8 = {0, 0, 0, 0, 0, 0, 0, 0};
    __builtin_amdgcn_tensor_load_to_lds(g0, g1, gz4, gz4, gz8, 0);
}

__device__ __forceinline__ void tdm_store_1d(void* gptr, const void* lptr, uint32_t n) {
    uint32x4 g0; int32x8 g1;
    tdm_desc_1d((uint64_t)(uintptr_t)gptr, (uint32_t)(uintptr_t)lptr, n, g0, g1);
    int32x4 gz4 = {0, 0, 0, 0};
    int32x8 gz8 = {0, 0, 0, 0, 0, 0, 0, 0};
    __builtin_amdgcn_tensor_store_from_lds(g0, g1, gz4, gz4, gz8, 0);
}

// -----------------------------------------------------------------------------
// K1: chunk-local partials with zero carry-in.
//   forward:  f_t = w*f_{t-1} + (e_t + b)          (t==0 exact: e + ln_pi)
//   backward: u_t = w*u_{t+1} + (e_t + b)          (t==T-1 exact: 2e + ln_pi)
// One wave per (b, chunk); lane = state; obvs chunk DMA'd to LDS by the TDM.
// -----------------------------------------------------------------------------
__global__ __launch_bounds__(32) void hmm_partials(
    const float* __restrict__ obvs, const float* __restrict__ ln_pi,
    const float* __restrict__ trans_w, const float* __restrict__ trans_b,
    const float* __restrict__ mu_p, const float* __restrict__ log_sigma,
    float* __restrict__ partF, float* __restrict__ partU) {
    __shared__ float lds_x[L_CHUNK];

    const int lane = threadIdx.x;
    const int bc = blockIdx.x;
    const int b = bc / NCHUNK, c = bc % NCHUNK;
    const int t0 = c * L_CHUNK;

    tdm_load_1d(obvs + (size_t)b * T_DIM + t0, &lds_x[0], L_CHUNK);

    const float w   = trans_w[lane];
    const float bi  = trans_b[lane];
    const float pi  = ln_pi[lane];
    const float mu  = mu_p[lane];
    const float ls  = log_sigma[lane];
    const float inv = __expf(-ls);

    __builtin_amdgcn_s_wait_tensorcnt((short)0);   // LDS tile ready

    float f = 0.0f;
    for (int i = 0; i < L_CHUNK; ++i) {
        float e = emit_ll(lds_x[i], mu, inv, ls);   // uniform broadcast DS read
        int t = t0 + i;
        f = (t == 0) ? (e + pi) : fmaf(w, f, e + bi);
    }

    float u = 0.0f;
    for (int i = L_CHUNK - 1; i >= 0; --i) {
        float e = emit_ll(lds_x[i], mu, inv, ls);
        int t = t0 + i;
        u = (t == T_DIM - 1) ? (2.0f * e + pi) : fmaf(w, u, e + bi);
    }

    partF[(size_t)bc * S_DIM + lane] = f;
    partU[(size_t)bc * S_DIM + lane] = u;
}

// -----------------------------------------------------------------------------
// K2: inter-chunk scan (32 steps) with coefficient w^L; rewrites part arrays
// in place so that part[b][c][s] becomes the carry-IN for chunk c.
// One wave per batch row b; lane = state.
// -----------------------------------------------------------------------------
__global__ __launch_bounds__(32) void hmm_scan(
    const float* __restrict__ trans_w,
    float* __restrict__ partF, float* __restrict__ partU) {
    const int lane = threadIdx.x;
    const int b = blockIdx.x;

    float wL = trans_w[lane];
#pragma unroll
    for (int i = 0; i < 7; ++i) wL *= wL;   // w^128 (L_CHUNK = 2^7)

    float* pF = partF + (size_t)b * NCHUNK * S_DIM + lane;
    float carry = 0.0f;
    for (int c = 0; c < NCHUNK; ++c) {
        float p = pF[(size_t)c * S_DIM];
        pF[(size_t)c * S_DIM] = carry;                  // carry-in (unused for c==0)
        carry = (c == 0) ? p : fmaf(wL, carry, p);      // chunk 0 partial is exact
    }

    float* pU = partU + (size_t)b * NCHUNK * S_DIM + lane;
    carry = 0.0f;
    for (int c = NCHUNK - 1; c >= 0; --c) {
        float p = pU[(size_t)c * S_DIM];
        pU[(size_t)c * S_DIM] = carry;                  // carry-in from the right (unused c==NC-1)
        carry = (c == NCHUNK - 1) ? p : fmaf(wL, carry, p);
    }
}

// -----------------------------------------------------------------------------
// K3: per-chunk backward fill (LDS) + forward combine + logsumexp.
// obvs chunk DMA'd in by the TDM; the finished 16 KB (L x S) result tile is
// DMA'd back out by TENSOR_STORE_FROM_LDS (one instruction instead of 128
// vector stores). One wave per (b, chunk); lane = state.
// -----------------------------------------------------------------------------
__global__ __launch_bounds__(32) void hmm_combine(
    const float* __restrict__ obvs, const float* __restrict__ ln_pi,
    const float* __restrict__ trans_w, const float* __restrict__ trans_b,
    const float* __restrict__ mu_p, const float* __restrict__ log_sigma,
    const float* __restrict__ partF, const float* __restrict__ partU,
    float* __restrict__ out) {
    __shared__ float lds_x[L_CHUNK];
    __shared__ float lds_g[L_CHUNK * S_DIM];   // 16 KB: b_t, then normalized gamma

    const int lane = threadIdx.x;
    const int bc = blockIdx.x;
    const int b = bc / NCHUNK, c = bc % NCHUNK;
    const int t0 = c * L_CHUNK;

    tdm_load_1d(obvs + (size_t)b * T_DIM + t0, &lds_x[0], L_CHUNK);

    const float w   = trans_w[lane];
    const float bi  = trans_b[lane];
    const float pi  = ln_pi[lane];
    const float mu  = mu_p[lane];
    const float ls  = log_sigma[lane];
    const float inv = __expf(-ls);

    const float carryF = partF[(size_t)bc * S_DIM + lane];
    const float carryU = partU[(size_t)bc * S_DIM + lane];

    __builtin_amdgcn_s_wait_tensorcnt((short)0);   // LDS tile ready

    // backward sweep: b_t = u_t - e_t, u_t = w*u_{t+1} + (e_t + bias)
    float u = carryU;
    for (int i = L_CHUNK - 1; i >= 0; --i) {
        int t = t0 + i;
        float e = emit_ll(lds_x[i], mu, inv, ls);
        u = (t == T_DIM - 1) ? (2.0f * e + pi) : fmaf(w, u, e + bi);
        lds_g[i * S_DIM + lane] = u - e;
    }

    // forward sweep + combine + logsumexp over the 32 states (wave lanes);
    // normalized result overwrites lds_g in place (same lane, same address).
    float f = carryF;
    for (int i = 0; i < L_CHUNK; ++i) {
        int t = t0 + i;
        float e = emit_ll(lds_x[i], mu, inv, ls);
        f = (t == 0) ? (e + pi) : fmaf(w, f, e + bi);
        float g = f + lds_g[i * S_DIM + lane];

        float m = g;
#pragma unroll
        for (int off = 16; off >= 1; off >>= 1)
            m = fmaxf(m, __shfl_xor(m, off, 32));
        float p = __expf(g - m);
#pragma unroll
        for (int off = 16; off >= 1; off >>= 1)
            p += __shfl_xor(p, off, 32);
        float lse = m + __logf(p);

        lds_g[i * S_DIM + lane] = g - lse;
    }

    // Fence this wave's LDS writes, then DMA the whole contiguous tile
    // out[(b*T + t0)*S .. +L*S) in one TDM store.
    asm volatile("s_wait_dscnt 0x0" ::: "memory");
    tdm_store_1d(out + ((size_t)b * T_DIM + t0) * S_DIM, &lds_g[0], L_CHUNK * S_DIM);
    __builtin_amdgcn_s_wait_tensorcnt((short)0);   // (S_ENDPGM also waits-idle)
}

// -----------------------------------------------------------------------------
extern "C" void kernel_launch(void* const* d_in, const int* in_sizes, int n_in,
                              void* d_out, int out_size, void* d_ws, size_t ws_size,
                              hipStream_t stream) {
    const float* obvs      = (const float*)d_in[0];
    const float* ln_pi     = (const float*)d_in[1];
    const float* trans_w   = (const float*)d_in[2];
    const float* trans_b   = (const float*)d_in[3];
    const float* mu        = (const float*)d_in[4];
    const float* log_sigma = (const float*)d_in[5];
    float* out = (float*)d_out;

    float* partF = (float*)d_ws;                                  // B*NC*S floats (1 MB)
    float* partU = partF + (size_t)B_DIM * NCHUNK * S_DIM;        // B*NC*S floats (1 MB)

    hmm_partials<<<B_DIM * NCHUNK, 32, 0, stream>>>(
        obvs, ln_pi, trans_w, trans_b, mu, log_sigma, partF, partU);
    hmm_scan<<<B_DIM, 32, 0, stream>>>(trans_w, partF, partU);
    hmm_combine<<<B_DIM * NCHUNK, 32, 0, stream>>>(
        obvs, ln_pi, trans_w, trans_b, mu, log_sigma, partF, partU, out);
}